// res_rand_GAE_70214125355147
// MI455X (gfx1250) — compile-verified
//
#include <hip/hip_runtime.h>
#include <hip/hip_bf16.h>

typedef __attribute__((ext_vector_type(2))) float v2f;
typedef __attribute__((ext_vector_type(8))) float v8f;

// ---------------------------------------------------------------------------
// Degree / normalization kernels
// ---------------------------------------------------------------------------
__global__ __launch_bounds__(256) void deg_init_kernel(float* __restrict__ deg, int n) {
    int i = blockIdx.x * 256 + threadIdx.x;
    if (i < n) deg[i] = 2.0f;  // two self loops per node
}

__global__ __launch_bounds__(256) void deg_edge_kernel(const int* __restrict__ dst,
                                                       float* __restrict__ deg, int e) {
    int i = blockIdx.x * 256 + threadIdx.x;
    if (i < e) atomicAdd(&deg[dst[i]], 1.0f);
}

__global__ __launch_bounds__(256) void dinv_kernel(float* __restrict__ d, int n) {
    int i = blockIdx.x * 256 + threadIdx.x;
    if (i < n) d[i] = rsqrtf(d[i]);  // deg >= 2 always
}

// ---------------------------------------------------------------------------
// Aggregation over 128-dim features: agg = Ahat * X
//   self part:  agg[i] = 2 * dinv[i]^2 * X[i]
//   edge part:  agg[dst] += X[src] * dinv[src]*dinv[dst]
// One wave (32 lanes) covers the 128 floats (float4 per lane) of one node/edge.
// ---------------------------------------------------------------------------
__global__ __launch_bounds__(256) void agg_self_kernel(const float* __restrict__ X,
                                                       const float* __restrict__ dinv,
                                                       float* __restrict__ agg, int n) {
    int t = blockIdx.x * 256 + threadIdx.x;
    if (t >= n * 32) return;
    int node = t >> 5;
    int fc = (t & 31) << 2;
    float di = dinv[node];
    float w = 2.0f * di * di;
    const float4 x = *reinterpret_cast<const float4*>(X + (size_t)node * 128 + fc);
    float4 o = make_float4(x.x * w, x.y * w, x.z * w, x.w * w);
    *reinterpret_cast<float4*>(agg + (size_t)node * 128 + fc) = o;
}

__global__ __launch_bounds__(256) void agg_edge_kernel(const int* __restrict__ src,
                                                       const int* __restrict__ dst,
                                                       const float* __restrict__ X,
                                                       const float* __restrict__ dinv,
                                                       float* __restrict__ agg, int e) {
    int t = blockIdx.x * 256 + threadIdx.x;
    if (t >= e * 32) return;
    int ed = t >> 5;
    int fc = (t & 31) << 2;
    int s = src[ed];
    int d = dst[ed];
    float w = dinv[s] * dinv[d];
    const float4 x = *reinterpret_cast<const float4*>(X + (size_t)s * 128 + fc);
    float* ap = agg + (size_t)d * 128 + fc;
    atomicAdd(ap + 0, x.x * w);
    atomicAdd(ap + 1, x.y * w);
    atomicAdd(ap + 2, x.z * w);
    atomicAdd(ap + 3, x.w * w);
}

// ---------------------------------------------------------------------------
// FP32 WMMA GEMM: out[M,Nc] = epilogue(A[M,K] @ W[K,Nc])
// Each wave computes a 16x64 tile (4 x v8f accumulators), K stepped by 4 using
// V_WMMA_F32_16X16X4_F32.  Fragment layouts per CDNA5 ISA 7.12.2:
//   A 16x4 : lanes 0-15 row M=lane, {K, K+1}; lanes 16-31 row M=lane-16, {K+2, K+3}
//   B 4x16 : lanes 0-15 col N=lane, rows {K, K+1}; lanes 16-31 rows {K+2, K+3}
//   C/D    : vgpr v, lanes 0-15 -> M=v, lanes 16-31 -> M=v+8, N = lane&15
// EPI: 0 = relu(acc + bias)        (GCN1 post / fc1)
//      1 = acc                     (h2 = x1 @ W2, bias+relu after aggregation)
//      2 = acc + bias + relu(aux + bias2)   (residual merge, written to d_out)
// ---------------------------------------------------------------------------
template <int EPI>
__global__ __launch_bounds__(256) void gemm_wmma_f32(const float* __restrict__ A,
                                                     const float* __restrict__ W,
                                                     const float* __restrict__ bias,
                                                     const float* __restrict__ bias2,
                                                     const float* __restrict__ aux,
                                                     float* __restrict__ out,
                                                     int M, int K, int Nc) {
    const int wid = threadIdx.x >> 5;
    const int lane = threadIdx.x & 31;
    const int tilesN = Nc >> 6;                    // 64-wide tiles
    const int totalWaves = (M >> 4) * tilesN;
    const int gw = blockIdx.x * 8 + wid;
    if (gw >= totalWaves) return;                  // wave-uniform: EXEC all-1s inside

    const int tm = gw / tilesN;
    const int tn = gw - tm * tilesN;
    const int row0 = tm << 4;
    const int col0 = tn << 6;
    const int r = lane & 15;
    const int half = lane >> 4;

    const float* __restrict__ Arow = A + (size_t)(row0 + r) * K;
    const int colb = col0 + r;

    v8f acc0 = {0.f, 0.f, 0.f, 0.f, 0.f, 0.f, 0.f, 0.f};
    v8f acc1 = acc0, acc2 = acc0, acc3 = acc0;

    for (int k = 0; k < K; k += 4) {
        const int kb = k + 2 * half;
        v2f a;
        a.x = Arow[kb];
        a.y = Arow[kb + 1];
        const float* __restrict__ Wr0 = W + (size_t)kb * Nc + colb;
        const float* __restrict__ Wr1 = Wr0 + Nc;
        v2f b0; b0.x = Wr0[0];  b0.y = Wr1[0];
        v2f b1; b1.x = Wr0[16]; b1.y = Wr1[16];
        v2f b2; b2.x = Wr0[32]; b2.y = Wr1[32];
        v2f b3; b3.x = Wr0[48]; b3.y = Wr1[48];
        acc0 = __builtin_amdgcn_wmma_f32_16x16x4_f32(false, a, false, b0, (short)0, acc0, false, false);
        acc1 = __builtin_amdgcn_wmma_f32_16x16x4_f32(false, a, false, b1, (short)0, acc1, false, false);
        acc2 = __builtin_amdgcn_wmma_f32_16x16x4_f32(false, a, false, b2, (short)0, acc2, false, false);
        acc3 = __builtin_amdgcn_wmma_f32_16x16x4_f32(false, a, false, b3, (short)0, acc3, false, false);
    }

#pragma unroll
    for (int v = 0; v < 8; ++v) {
        const int orow = row0 + v + 8 * half;
        float* __restrict__ orw = out + (size_t)orow * Nc;
        const float vals[4] = {acc0[v], acc1[v], acc2[v], acc3[v]};
#pragma unroll
        for (int sub = 0; sub < 4; ++sub) {
            const int col = colb + sub * 16;
            float x = vals[sub];
            if (EPI == 0) {
                x = fmaxf(x + bias[col], 0.0f);
            } else if (EPI == 2) {
                x = x + bias[col] + fmaxf(aux[(size_t)orow * Nc + col] + bias2[col], 0.0f);
            }
            orw[col] = x;
        }
    }
}

// ---------------------------------------------------------------------------
// Final head: out[node] = F1[node, 0:256] . Wfc2 + bfc2   (one wave per node)
// ---------------------------------------------------------------------------
__global__ __launch_bounds__(256) void fc2_kernel(const float* __restrict__ F1,
                                                  const float* __restrict__ w,
                                                  const float* __restrict__ b,
                                                  float* __restrict__ out, int m) {
    int wid = threadIdx.x >> 5;
    int lane = threadIdx.x & 31;
    int node = blockIdx.x * 8 + wid;
    if (node >= m) return;
    const float* __restrict__ f = F1 + (size_t)node * 256;
    float s = 0.0f;
#pragma unroll
    for (int j = 0; j < 8; ++j) s += f[lane + 32 * j] * w[lane + 32 * j];
#pragma unroll
    for (int off = 16; off > 0; off >>= 1) s += __shfl_down(s, off, 32);
    if (lane == 0) out[node] = s + b[0];
}

// ---------------------------------------------------------------------------
// Launch
// ---------------------------------------------------------------------------
static inline int gemm_blocks(int M, int Nc) {
    int totalWaves = (M >> 4) * (Nc >> 6);
    return (totalWaves + 7) / 8;
}

extern "C" void kernel_launch(void* const* d_in, const int* in_sizes, int n_in,
                              void* d_out, int out_size, void* d_ws, size_t ws_size,
                              hipStream_t stream) {
    const float* node_emb = (const float*)d_in[0];
    const int* ei = (const int*)d_in[1];
    const float* W1 = (const float*)d_in[2];
    const float* b1 = (const float*)d_in[3];
    const float* W2 = (const float*)d_in[4];
    const float* b2 = (const float*)d_in[5];
    const float* Wres = (const float*)d_in[6];
    const float* bres = (const float*)d_in[7];
    const float* Wfc1 = (const float*)d_in[8];
    const float* bfc1 = (const float*)d_in[9];
    const float* Wfc2 = (const float*)d_in[10];
    const float* bfc2 = (const float*)d_in[11];

    const int Nn = in_sizes[0] / 128;
    const int E = in_sizes[1] / 2;
    const int* src = ei;
    const int* dst = ei + E;

    // workspace layout (floats): dinv[N] | tmpB[N*128] | tmpC[N*128] | tmpA[N*512]
    float* ws = (float*)d_ws;
    float* dinv = ws;
    float* tmpB = dinv + Nn;                 // agg0, later h2
    float* tmpC = tmpB + (size_t)Nn * 128;   // agg2
    float* tmpA = tmpC + (size_t)Nn * 128;   // x1, later F1 (N*256 <= N*512)

    float* xout = (float*)d_out;             // [N,128]
    float* Aout = xout + (size_t)Nn * 128;   // [N,1]

    const int nb_n = (Nn + 255) / 256;
    const int nb_e = (E + 255) / 256;
    const int nb_nf = (Nn * 32 + 255) / 256;
    const int nb_ef = (E * 32 + 255) / 256;

    // 1) deg / dinv
    deg_init_kernel<<<nb_n, 256, 0, stream>>>(dinv, Nn);
    deg_edge_kernel<<<nb_e, 256, 0, stream>>>(dst, dinv, E);
    dinv_kernel<<<nb_n, 256, 0, stream>>>(dinv, Nn);

    // 2) agg0 = Ahat * X   (aggregate BEFORE GEMM: 128-d messages, not 512-d)
    agg_self_kernel<<<nb_nf, 256, 0, stream>>>(node_emb, dinv, tmpB, Nn);
    agg_edge_kernel<<<nb_ef, 256, 0, stream>>>(src, dst, node_emb, dinv, tmpB, E);

    // 3) x1 = relu(agg0 @ W1 + b1)  [N,512]
    gemm_wmma_f32<0><<<gemm_blocks(Nn, 512), 256, 0, stream>>>(
        tmpB, W1, b1, nullptr, nullptr, tmpA, Nn, 128, 512);

    // 4) h2 = x1 @ W2  [N,128]
    gemm_wmma_f32<1><<<gemm_blocks(Nn, 128), 256, 0, stream>>>(
        tmpA, W2, nullptr, nullptr, nullptr, tmpB, Nn, 512, 128);

    // 5) agg2 = Ahat * h2
    agg_self_kernel<<<nb_nf, 256, 0, stream>>>(tmpB, dinv, tmpC, Nn);
    agg_edge_kernel<<<nb_ef, 256, 0, stream>>>(src, dst, tmpB, dinv, tmpC, E);

    // 6) x = relu(agg2 + b2) + x1 @ Wres + bres  -> d_out[0 : N*128]
    gemm_wmma_f32<2><<<gemm_blocks(Nn, 128), 256, 0, stream>>>(
        tmpA, Wres, bres, b2, tmpC, xout, Nn, 512, 128);

    // 7) F1 = relu(x @ Wfc1 + bfc1)  [N,256]
    gemm_wmma_f32<0><<<gemm_blocks(Nn, 256), 256, 0, stream>>>(
        xout, Wfc1, bfc1, nullptr, nullptr, tmpA, Nn, 128, 256);

    // 8) A = F1 @ Wfc2 + bfc2 -> d_out[N*128 : N*128 + N]
    fc2_kernel<<<(Nn + 7) / 8, 256, 0, stream>>>(tmpA, Wfc2, bfc2, Aout, Nn);
}